// BLoss_37890201485738
// MI455X (gfx1250) — compile-verified
//
#include <hip/hip_runtime.h>

// ---------------------------------------------------------------------------
// loss = mean_i( max(0, sum_j B[i,j] - 1) ),  B: 8192 x 16384 fp32
// Bandwidth-bound (512 MiB read once -> ~23 us at 23.3 TB/s).
// Pass 1: one block per row, NT b128 streaming loads, per-row hinge -> d_ws.
// Pass 2: one block reduces 8192 hinges -> scalar mean.
// Cross-lane (32-lane) reduction done exactly in fp32 via chained
// V_WMMA_F32_16X16X4_F32 with ones-matrices (CDNA5 wave32 WMMA path).
// Deterministic: fixed reduction tree, no atomics.
// ---------------------------------------------------------------------------

typedef __attribute__((ext_vector_type(2))) float v2f;
typedef __attribute__((ext_vector_type(4))) float v4f;
typedef __attribute__((ext_vector_type(8))) float v8f;

#define ROWS 8192
#define COLS 16384
#define TPB  256
#define V4_PER_ROW (COLS / 4)      // 4096
#define V4_PER_THREAD (V4_PER_ROW / TPB)  // 16

// Exact fp32 sum of p across all 32 lanes of the wave using WMMA.
// Step 1: D1[m][n] = p[m] + p[m+16]           (A = {p,0}, B = ones)
// Step 2: acc += ones x D1-pairs  (4x)  ->    every element = sum of all 16
//         row values of D1 = full 32-lane sum.
// Requires EXEC == all ones (callers have no divergence before this point).
__device__ __forceinline__ float wave32_sum_wmma(float p) {
  v2f a;    a[0] = p;    a[1] = 0.0f;
  v2f ones; ones[0] = 1.0f; ones[1] = 1.0f;
  v8f zero = {0.f, 0.f, 0.f, 0.f, 0.f, 0.f, 0.f, 0.f};

  v8f d1 = __builtin_amdgcn_wmma_f32_16x16x4_f32(
      false, a, false, ones, (short)0, zero, false, false);

  v8f acc = {0.f, 0.f, 0.f, 0.f, 0.f, 0.f, 0.f, 0.f};
#pragma unroll
  for (int i = 0; i < 4; ++i) {
    v2f b; b[0] = d1[2 * i]; b[1] = d1[2 * i + 1];
    acc = __builtin_amdgcn_wmma_f32_16x16x4_f32(
        false, ones, false, b, (short)0, acc, false, false);
  }
  return acc[0];
}

// ---- Pass 1: per-row sum + hinge -----------------------------------------
__global__ __launch_bounds__(TPB) void BLoss_rowsum_hinge(
    const float* __restrict__ B, float* __restrict__ hinge) {
  const int row = blockIdx.x;
  const int t   = threadIdx.x;

  const v4f* rp = reinterpret_cast<const v4f*>(B) + (size_t)row * V4_PER_ROW;

  v4f acc = {0.f, 0.f, 0.f, 0.f};
#pragma unroll
  for (int i = 0; i < V4_PER_THREAD; ++i) {
    // Streaming read-once data: non-temporal so we don't thrash L2.
    v4f v = __builtin_nontemporal_load(&rp[t + TPB * i]);
    acc += v;
  }
  float p = (acc[0] + acc[1]) + (acc[2] + acc[3]);

  // 32-lane exact reduce via WMMA (EXEC all ones here).
  float wsum = wave32_sum_wmma(p);

  __shared__ float s[TPB / 32];
  const int wave = t >> 5;
  const int lane = t & 31;
  if (lane == 0) s[wave] = wsum;
  __syncthreads();

  if (t == 0) {
    float r = ((s[0] + s[1]) + (s[2] + s[3])) +
              ((s[4] + s[5]) + (s[6] + s[7]));
    float h = r - 1.0f;
    hinge[row] = h > 0.0f ? h : 0.0f;
  }
}

// ---- Pass 2: mean of 8192 hinge values -----------------------------------
__global__ __launch_bounds__(TPB) void BLoss_final_mean(
    const float* __restrict__ hinge, float* __restrict__ out) {
  const int t = threadIdx.x;

  float p = 0.0f;
#pragma unroll
  for (int i = 0; i < ROWS / TPB; ++i) {  // 32 coalesced strided loads
    p += hinge[t + TPB * i];
  }

  float wsum = wave32_sum_wmma(p);  // EXEC all ones

  __shared__ float s[TPB / 32];
  const int wave = t >> 5;
  const int lane = t & 31;
  if (lane == 0) s[wave] = wsum;
  __syncthreads();

  if (t == 0) {
    float r = ((s[0] + s[1]) + (s[2] + s[3])) +
              ((s[4] + s[5]) + (s[6] + s[7]));
    out[0] = r * (1.0f / (float)ROWS);  // PENALTY_B == 1.0
  }
}

extern "C" void kernel_launch(void* const* d_in, const int* in_sizes, int n_in,
                              void* d_out, int out_size, void* d_ws,
                              size_t ws_size, hipStream_t stream) {
  (void)in_sizes; (void)n_in; (void)out_size; (void)ws_size;

  const float* B   = (const float*)d_in[0];
  float*       out = (float*)d_out;
  float*       ws  = (float*)d_ws;   // 8192 floats of per-row hinge

  BLoss_rowsum_hinge<<<ROWS, TPB, 0, stream>>>(B, ws);
  BLoss_final_mean<<<1, TPB, 0, stream>>>(ws, out);
}